// ClusterAboveThreshold_86449101734211
// MI455X (gfx1250) — compile-verified
//
#include <hip/hip_runtime.h>
#include <hip/hip_bf16.h>
#include <stdint.h>

// Problem constants (match reference)
#define BB 8
#define HH 1024
#define WW 1024
#define NN (BB * HH * WW)          // 8,388,608
#define HWD (HH * WW)              // 1<<20
#define THRESH 1.5f
#define MIN_AREA 3

// ---------------------------------------------------------------------------
// Union-find helpers on global label array.
// Invariant: lab[p] == NN for background; for foreground, lab forms a forest
// where the root of each tree is the minimum flat index of the component.
// ---------------------------------------------------------------------------
__device__ __forceinline__ int find_root(int* lab, int x) {
    for (;;) {
        int p = lab[x];
        if (p == x) return x;
        int gp = lab[p];
        if (gp == p) return p;
        lab[x] = gp;   // path halving; benign race (always points to an ancestor)
        x = gp;
    }
}

__device__ __forceinline__ void merge_labels(int* lab, int a, int b) {
    a = find_root(lab, a);
    b = find_root(lab, b);
    while (a != b) {
        if (a > b) { int t = a; a = b; b = t; }   // keep a < b: link larger onto smaller
        int old = atomicMin(&lab[b], a);
        if (old == b) return;                      // we installed the link
        b = old;                                   // someone else linked b; walk up, retry
    }
}

// ---------------------------------------------------------------------------
// Kernel 1: per-tile local CCL in LDS + init of all workspace arrays.
// Block = 32x32 threads (32 wave32s), one tile per block.
// Input tile staged into LDS via the Tensor Data Mover (TDM).
// ---------------------------------------------------------------------------
typedef unsigned int tdm_v4u __attribute__((ext_vector_type(4)));
typedef int          tdm_v8i __attribute__((ext_vector_type(8)));
typedef int          tdm_v4i __attribute__((ext_vector_type(4)));

__global__ __launch_bounds__(1024) void tile_label_kernel(
    const float* __restrict__ x, int* __restrict__ lab,
    int* __restrict__ area, int* __restrict__ cmaxb, int* __restrict__ amin)
{
    __shared__ float tile[32 * 32];   // first LDS allocation -> offset 0 (TDM dest)
    __shared__ int   llab[32 * 32];
    __shared__ int   s_changed;

    const int tc = blockIdx.x;        // tile col (0..31)
    const int tr = blockIdx.y;        // tile row (0..31)
    const int b  = blockIdx.z;        // batch (0..7)
    const int tx = threadIdx.x;       // col within tile
    const int ty = threadIdx.y;       // row within tile (== wave id)
    const int l  = ty * 32 + tx;

    // ---- Stage the 32x32 float tile into LDS via TDM ----
#if __has_builtin(__builtin_amdgcn_tensor_load_to_lds)
    if (ty == 0) {  // wave-uniform branch: only wave 0 issues the TDM op
        uint64_t gaddr = (uint64_t)(uintptr_t)x +
                         ((uint64_t)((b * HH + tr * 32) * WW + tc * 32) << 2);
        // D# group0: count=1, lds_addr=0 (tile at LDS offset 0),
        //            global_addr[56:0] in bits 120:64, type=2 ("image")
        tdm_v4u g0 = { 1u,
                       0u,
                       (unsigned)(gaddr & 0xFFFFFFFFull),
                       (unsigned)(((gaddr >> 32) & 0x1FFFFFFull) | (2u << 30)) };
        // D# group1: data_size=2 (4B) @17:16; tensor_dim0=1024 @79:48;
        //            tensor_dim1=1024 @111:80; tile_dim0=32 @127:112;
        //            tile_dim1=32 @143:128; tensor_dim0_stride=1024 @207:160
        tdm_v8i g1 = { (int)0x00020000,
                       (int)(1024u << 16),
                       (int)(1024u << 16),
                       (int)(32u << 16),
                       32,
                       1024,
                       0, 0 };
        tdm_v4i gz = { 0, 0, 0, 0 };
#if defined(__clang_major__) && (__clang_major__ >= 23)
        tdm_v8i g4 = { 0, 0, 0, 0, 0, 0, 0, 0 };
        __builtin_amdgcn_tensor_load_to_lds(g0, g1, gz, gz, g4, 0);
#else
        __builtin_amdgcn_tensor_load_to_lds(g0, g1, gz, gz, 0);
#endif
        __builtin_amdgcn_s_wait_tensorcnt(0);
    }
#else
    // Fallback: plain coalesced load (one row per wave)
    tile[l] = x[(b * HH + tr * 32 + ty) * WW + tc * 32 + tx];
#endif
    __syncthreads();

    const float v  = tile[l];
    const bool  fg = (v >= THRESH);
    llab[l] = fg ? l : 1024;          // 1024 = local background sentinel
    if (l == 0) s_changed = 0;
    __syncthreads();

    // ---- Iterative min-label propagation (8-connectivity) inside the tile ----
    for (;;) {
        if (fg) {
            int cur = llab[l];
            int m = cur;
            if (tx > 0)  m = min(m, llab[l - 1]);
            if (tx < 31) m = min(m, llab[l + 1]);
            if (ty > 0) {
                m = min(m, llab[l - 32]);
                if (tx > 0)  m = min(m, llab[l - 33]);
                if (tx < 31) m = min(m, llab[l - 31]);
            }
            if (ty < 31) {
                m = min(m, llab[l + 32]);
                if (tx > 0)  m = min(m, llab[l + 31]);
                if (tx < 31) m = min(m, llab[l + 33]);
            }
            if (m < cur) { llab[l] = m; s_changed = 1; }
        }
        __syncthreads();
        if (s_changed == 0) break;
        __syncthreads();
        if (l == 0) s_changed = 0;
        __syncthreads();
    }

    // ---- Emit global labels (depth-1 trees rooted at tile-local min index) ----
    const int p = (b * HH + tr * 32 + ty) * WW + (tc * 32 + tx);
    int g = NN;
    if (fg) {
        int lr = llab[l];
        g = (b * HH + tr * 32 + (lr >> 5)) * WW + (tc * 32 + (lr & 31));
    }
    lab[p]   = g;
    area[p]  = 0;
    cmaxb[p] = 0;          // max of positive floats; 0 bits == 0.0f
    amin[p]  = HWD;        // sentinel (within-image index domain)
}

// ---------------------------------------------------------------------------
// Kernel 2: union across tile boundaries (forward neighbors only).
// Boundary test is pure index math and runs BEFORE any memory access, so
// ~91% of threads retire without touching HBM.
// ---------------------------------------------------------------------------
__global__ void border_merge_kernel(int* __restrict__ lab)
{
    const int p = blockIdx.x * blockDim.x + threadIdx.x;
    if (p >= NN) return;

    const int w = p & (WW - 1);
    const int h = (p >> 10) & (HH - 1);
    const bool re = ((w & 31) == 31);
    const bool le = ((w & 31) == 0);
    const bool be = ((h & 31) == 31);
    if (!(re | le | be)) return;                    // interior: fully merged locally
    if (lab[p] >= NN) return;                       // background

    if (re && w < WW - 1 && lab[p + 1] < NN) merge_labels(lab, p, p + 1);
    if (h < HH - 1) {
        if (be) {
            if (lab[p + WW] < NN)                      merge_labels(lab, p, p + WW);
            if (w > 0      && lab[p + WW - 1] < NN)    merge_labels(lab, p, p + WW - 1);
            if (w < WW - 1 && lab[p + WW + 1] < NN)    merge_labels(lab, p, p + WW + 1);
        } else {
            if (le && w > 0      && lab[p + WW - 1] < NN) merge_labels(lab, p, p + WW - 1);
            if (re && w < WW - 1 && lab[p + WW + 1] < NN) merge_labels(lab, p, p + WW + 1);
        }
    }
}

// ---------------------------------------------------------------------------
// Kernel 3: compress labels + per-component area and max.
// 4 pixels/thread: b128 loads of lab and x. atomicMax on float bits is valid
// because all foreground values are >= 1.5 > 0 (int order == float order).
// ---------------------------------------------------------------------------
__global__ void stats_kernel(const float* __restrict__ x, int* __restrict__ lab,
                             int* __restrict__ area, int* __restrict__ cmaxb)
{
    const int t = blockIdx.x * blockDim.x + threadIdx.x;
    const int p0 = t * 4;
    if (p0 >= NN) return;
    const int4   l4 = *(const int4*)(lab + p0);
    const float4 v4 = *(const float4*)(x + p0);
    const int   li[4] = { l4.x, l4.y, l4.z, l4.w };
    const float vi[4] = { v4.x, v4.y, v4.z, v4.w };
#pragma unroll
    for (int i = 0; i < 4; ++i) {
        if (li[i] < NN) {
            const int p = p0 + i;
            const int r = find_root(lab, p);
            lab[p] = r;                             // full compression
            atomicAdd(&area[r], 1);
            atomicMax(&cmaxb[r], __float_as_int(vi[i]));
        }
    }
}

// ---------------------------------------------------------------------------
// Kernel 4: first (min within-image index) occurrence of the component max.
// ---------------------------------------------------------------------------
__global__ void argmax_kernel(const float* __restrict__ x, const int* __restrict__ lab,
                              const int* __restrict__ cmaxb, int* __restrict__ amin)
{
    const int t = blockIdx.x * blockDim.x + threadIdx.x;
    const int p0 = t * 4;
    if (p0 >= NN) return;
    const int4   l4 = *(const int4*)(lab + p0);
    const float4 v4 = *(const float4*)(x + p0);
    const int   li[4] = { l4.x, l4.y, l4.z, l4.w };
    const float vi[4] = { v4.x, v4.y, v4.z, v4.w };
#pragma unroll
    for (int i = 0; i < 4; ++i) {
        const int r = li[i];
        if (r < NN && __float_as_int(vi[i]) == cmaxb[r])
            atomicMin(&amin[r], (p0 + i) & (HWD - 1));   // within-image flat index
    }
}

// ---------------------------------------------------------------------------
// Kernel 5: dense output rewrite: (max, row, col) planes, b128 stores.
// row/col emitted as floats (exact for -1..1023); planes concatenated.
// ---------------------------------------------------------------------------
__global__ void output_kernel(const int* __restrict__ lab, const int* __restrict__ area,
                              const int* __restrict__ cmaxb, const int* __restrict__ amin,
                              float* __restrict__ out0, float* __restrict__ out1,
                              float* __restrict__ out2)
{
    const int t = blockIdx.x * blockDim.x + threadIdx.x;
    const int p0 = t * 4;
    if (p0 >= NN) return;
    const int4 l4 = *(const int4*)(lab + p0);
    const int  li[4] = { l4.x, l4.y, l4.z, l4.w };
    float o0[4], o1[4], o2[4];
#pragma unroll
    for (int i = 0; i < 4; ++i) {
        const int r = li[i];
        float om = 0.0f, orow = -1.0f, ocol = -1.0f;
        if (r < NN && area[r] > MIN_AREA) {
            om = __int_as_float(cmaxb[r]);
            const int a = amin[r];
            orow = (float)(a >> 10);                // a / W
            ocol = (float)(a & (WW - 1));           // a % W
        }
        o0[i] = om; o1[i] = orow; o2[i] = ocol;
    }
    *(float4*)(out0 + p0) = make_float4(o0[0], o0[1], o0[2], o0[3]);
    *(float4*)(out1 + p0) = make_float4(o1[0], o1[1], o1[2], o1[3]);
    *(float4*)(out2 + p0) = make_float4(o2[0], o2[1], o2[2], o2[3]);
}

// ---------------------------------------------------------------------------
extern "C" void kernel_launch(void* const* d_in, const int* in_sizes, int n_in,
                              void* d_out, int out_size, void* d_ws, size_t ws_size,
                              hipStream_t stream)
{
    const float* x = (const float*)d_in[0];
    float* out = (float*)d_out;

    int* lab   = (int*)d_ws;           // NN ints
    int* area  = lab   + NN;           // NN ints
    int* cmaxb = area  + NN;           // NN ints (float bits)
    int* amin  = cmaxb + NN;           // NN ints   (total 128 MB)

    dim3 blk1(32, 32, 1);
    dim3 grd1(WW / 32, HH / 32, BB);
    tile_label_kernel<<<grd1, blk1, 0, stream>>>(x, lab, area, cmaxb, amin);

    const int T = 256;
    border_merge_kernel<<<NN / T, T, 0, stream>>>(lab);
    const int G4 = NN / 4 / T;
    stats_kernel<<<G4, T, 0, stream>>>(x, lab, area, cmaxb);
    argmax_kernel<<<G4, T, 0, stream>>>(x, lab, cmaxb, amin);
    output_kernel<<<G4, T, 0, stream>>>(lab, area, cmaxb, amin,
                                        out, out + NN, out + 2 * NN);
}